// KalmanSmoother1D_78649441124424
// MI455X (gfx1250) — compile-verified
//
#include <hip/hip_runtime.h>

#define MINVARF 1e-6f
#define TILE 4096
#define NT 256
#define PER (TILE / NT)
#define NRED 512

// ---------------- CDNA5 TDM staging (async tensor load to LDS) ----------------
#if defined(__HIP_DEVICE_COMPILE__) && defined(__has_builtin)
#if __has_builtin(__builtin_amdgcn_tensor_load_to_lds) && __has_builtin(__builtin_amdgcn_s_wait_tensorcnt)
#define USE_TDM 1
#endif
#endif
#ifndef USE_TDM
#define USE_TDM 0
#endif

typedef unsigned int u32x4 __attribute__((ext_vector_type(4)));
typedef int i32x8 __attribute__((ext_vector_type(8)));
typedef int i32x4 __attribute__((ext_vector_type(4)));

// Stage nelem contiguous floats from global into LDS. TDM path: one wave-level
// DMA issued by wave 0 (EXEC ignored by TENSOR_LOAD_TO_LDS); descriptor per
// CDNA5 ISA 8.3/8.4: count=1, type=2, data_size=4B, tile = 1 row x nelem.
// Toolchain uses the 6-arg builtin: (g0 u32x4, g1 i32x8, g2 i32x4, g3 i32x4, g4 i32x8, cpol).
__device__ __forceinline__ void stage_to_lds(const float* gsrc, float* ldst, int nelem) {
#if USE_TDM
  if (threadIdx.x < 32u) {
    unsigned long long ga = (unsigned long long)(const void*)gsrc;
    unsigned lds_addr = (unsigned)(unsigned long long)(void*)ldst;  // low 32 bits of flat = LDS offset
    unsigned td0 = (unsigned)nelem;
    u32x4 g0;
    g0.x = 1u;                                                   // count = 1 valid descriptor
    g0.y = lds_addr;                                             // lds_addr
    g0.z = (unsigned)(ga & 0xffffffffull);                       // global_addr[31:0]
    g0.w = (unsigned)((ga >> 32) & 0x01ffffffull) | (2u << 30);  // global_addr[56:32], type=2
    i32x8 g1;
    g1[0] = (int)(2u << 16);                                     // data_size = 4 bytes
    g1[1] = (int)((td0 & 0xffffu) << 16);                        // tensor_dim0[15:0]
    g1[2] = (int)(((td0 >> 16) & 0xffffu) | (1u << 16));         // tensor_dim0[31:16], tensor_dim1=1
    g1[3] = (int)((td0 & 0xffffu) << 16);                        // tensor_dim1 hi=0, tile_dim0 = nelem
    g1[4] = (int)1u;                                             // tile_dim1 = 1, tile_dim2 = 0
    g1[5] = (int)td0;                                            // tensor_dim0_stride (lo)
    g1[6] = 0;
    g1[7] = 0;
    i32x4 gz4 = {0, 0, 0, 0};
    i32x8 gz8 = {0, 0, 0, 0, 0, 0, 0, 0};
    __builtin_amdgcn_tensor_load_to_lds(g0, g1, gz4, gz4, gz8, 0);
  }
#else
  for (int i = (int)threadIdx.x; i < nelem; i += (int)blockDim.x) ldst[i] = gsrc[i];
#endif
}

__device__ __forceinline__ void stage_wait() {
#if USE_TDM
  if (threadIdx.x < 32u) __builtin_amdgcn_s_wait_tensorcnt(0);
#endif
  __syncthreads();
}

// ---------------- Moebius (Riccati) and affine scan algebra ----------------
struct Mob { float a, b, c, d; };          // P -> (aP+b)/(cP+d)
__device__ __forceinline__ Mob mob_id() { Mob m; m.a = 1.f; m.b = 0.f; m.c = 0.f; m.d = 1.f; return m; }
__device__ __forceinline__ Mob mob_mul(Mob x, Mob w) {  // x applied after w => x*w, normalized
  Mob r;
  r.a = x.a * w.a + x.b * w.c;
  r.b = x.a * w.b + x.b * w.d;
  r.c = x.c * w.a + x.d * w.c;
  r.d = x.c * w.b + x.d * w.d;
  float s = 1.f / (fabsf(r.a) + fabsf(r.b) + fabsf(r.c) + fabsf(r.d) + 1e-30f);
  r.a *= s; r.b *= s; r.c *= s; r.d *= s;
  return r;
}
__device__ __forceinline__ Mob mob_step(float q, float rr) {  // P_filt = r(P+q)/(P+q+r)
  Mob m; m.a = rr; m.b = rr * q; m.c = 1.f; m.d = q + rr; return m;
}
__device__ __forceinline__ float mob_apply(Mob m, float p) { return (m.a * p + m.b) / (m.c * p + m.d); }
__device__ __forceinline__ void mob_st(float* s, int k, Mob m) {
  float* p = s + 4 * k; p[0] = m.a; p[1] = m.b; p[2] = m.c; p[3] = m.d;
}
__device__ __forceinline__ Mob mob_ld(const float* s, int k) {
  const float* p = s + 4 * k; Mob m; m.a = p[0]; m.b = p[1]; m.c = p[2]; m.d = p[3]; return m;
}

struct Aff { float a, b; };                // m -> a*m + b
__device__ __forceinline__ Aff aff_id() { Aff f; f.a = 1.f; f.b = 0.f; return f; }
__device__ __forceinline__ Aff aff_comp(Aff late, Aff early) {  // late applied after early
  Aff r; r.a = late.a * early.a; r.b = late.a * early.b + late.b; return r;
}
__device__ __forceinline__ void aff_st(float* s, int k, Aff f) { s[2 * k] = f.a; s[2 * k + 1] = f.b; }
__device__ __forceinline__ Aff aff_ld(const float* s, int k) { Aff f; f.a = s[2 * k]; f.b = s[2 * k + 1]; return f; }

__device__ __forceinline__ float ld_gs(const float* g, const float* s_arr, int lo, int hi, int idx) {
  return (idx >= lo && idx < hi) ? s_arr[idx - lo] : g[idx];
}

// LDS Hillis-Steele scans across NT threads (wave-size agnostic, wave32-safe)
__device__ __forceinline__ void mob_scan_inc(float* sS, int k, Mob v) {
  mob_st(sS, k, v);
  __syncthreads();
  for (int off = 1; off < NT; off <<= 1) {
    Mob mine = mob_ld(sS, k);
    Mob prev = mob_id();
    bool have = (k >= off);
    if (have) prev = mob_ld(sS, k - off);
    __syncthreads();
    if (have) mob_st(sS, k, mob_mul(mine, prev));
    __syncthreads();
  }
}
__device__ __forceinline__ void aff_scan_inc(float* sS, int k, Aff v) {
  aff_st(sS, k, v);
  __syncthreads();
  for (int off = 1; off < NT; off <<= 1) {
    Aff mine = aff_ld(sS, k);
    Aff prev = aff_id();
    bool have = (k >= off);
    if (have) prev = aff_ld(sS, k - off);
    __syncthreads();
    if (have) aff_st(sS, k, aff_comp(mine, prev));
    __syncthreads();
  }
}
__device__ __forceinline__ void aff_scan_rev(float* sS, int k, Aff v) {  // suffix scan
  aff_st(sS, k, v);
  __syncthreads();
  for (int off = 1; off < NT; off <<= 1) {
    Aff mine = aff_ld(sS, k);
    Aff nxt = aff_id();
    bool have = (k + off < NT);
    if (have) nxt = aff_ld(sS, k + off);
    __syncthreads();
    if (have) aff_st(sS, k, aff_comp(mine, nxt));
    __syncthreads();
  }
}

__device__ __forceinline__ Mob thread_mob_agg(const float* times, const float* sT,
                                              int lo, int hi, int s, int e, float qb, float rr) {
  Mob acc = mob_id();
  if (s < hi) {
    float tprev = 0.f;
    if (s > 0) tprev = ld_gs(times, sT, lo, hi, s - 1);
    for (int t = s; t < e; ++t) {
      float tc = sT[t - lo];
      float dtv = (t == 0) ? 1.0f : fmaxf(tc - tprev, MINVARF);
      tprev = tc;
      acc = mob_mul(mob_step(qb * dtv, rr), acc);
    }
  }
  return acc;
}

// ---------------- K1/K2: variance reduction -> {q_base, r, p0} ----------------
__global__ void ks_var_partial(const float* __restrict__ v, int n, double* __restrict__ part) {
  __shared__ double sh1[NT];
  __shared__ double sh2[NT];
  double s = 0.0, s2 = 0.0;
  const int stride = NT * NRED;
  for (int i = (int)(blockIdx.x * NT + threadIdx.x); i < n; i += stride) {
    float x = v[i];
    __builtin_prefetch(v + i + stride, 0, 0);  // global_prefetch_b8
    s += (double)x;
    s2 += (double)x * (double)x;
  }
  int k = (int)threadIdx.x;
  sh1[k] = s; sh2[k] = s2;
  __syncthreads();
  for (int off = NT / 2; off > 0; off >>= 1) {
    if (k < off) { sh1[k] += sh1[k + off]; sh2[k] += sh2[k + off]; }
    __syncthreads();
  }
  if (k == 0) { part[2 * blockIdx.x] = sh1[0]; part[2 * blockIdx.x + 1] = sh2[0]; }
}

__global__ void ks_var_final(const double* __restrict__ part, int nred, int n, float* __restrict__ scal) {
  __shared__ double sh1[NT];
  __shared__ double sh2[NT];
  double s = 0.0, s2 = 0.0;
  for (int i = (int)threadIdx.x; i < nred; i += NT) { s += part[2 * i]; s2 += part[2 * i + 1]; }
  int k = (int)threadIdx.x;
  sh1[k] = s; sh2[k] = s2;
  __syncthreads();
  for (int off = NT / 2; off > 0; off >>= 1) {
    if (k < off) { sh1[k] += sh1[k + off]; sh2[k] += sh2[k + off]; }
    __syncthreads();
  }
  if (k == 0) {
    double mean = sh1[0] / (double)n;
    double var = sh2[0] / (double)n - mean * mean;
    if (var < 0.0) var = 0.0;
    float var_y = (float)var + MINVARF;   // var(values) + MIN_VAR
    scal[0] = 0.05f * var_y;              // q_base
    scal[1] = var_y + MINVARF;            // r
    scal[2] = var_y + MINVARF;            // p0
  }
}

// ---------------- K3: per-block Moebius aggregates for P ----------------
__global__ void ks_fwdP_agg(const float* __restrict__ times, int n,
                            const float* __restrict__ scal, float* __restrict__ blkM) {
  extern __shared__ float sm[];
  float* sT = sm;            // TILE
  float* sS = sm + TILE;     // NT*4
  int lo = (int)blockIdx.x * TILE;
  int hi = lo + TILE; if (hi > n) hi = n;
  stage_to_lds(times + lo, sT, hi - lo);
  stage_wait();
  float qb = scal[0], rr = scal[1];
  int k = (int)threadIdx.x;
  int s = lo + k * PER;
  int e = s + PER; if (e > hi) e = hi;
  Mob acc = thread_mob_agg(times, sT, lo, hi, s, e, qb, rr);
  mob_scan_inc(sS, k, acc);
  if (k == NT - 1) mob_st(blkM, (int)blockIdx.x, mob_ld(sS, k));
}

// ---------------- K4: P carry-in per block (single 256-thread block) ----------------
__global__ void ks_carryP(const float* __restrict__ blkM, int nb,
                          const float* __restrict__ scal, float* __restrict__ pcar) {
  __shared__ float sS[NT * 4];
  int k = (int)threadIdx.x;
  int g = (nb + NT - 1) / NT;
  Mob acc = mob_id();
  for (int i = 0; i < g; ++i) { int b = k * g + i; if (b < nb) acc = mob_mul(mob_ld(blkM, b), acc); }
  mob_scan_inc(sS, k, acc);
  Mob excl = (k == 0) ? mob_id() : mob_ld(sS, k - 1);
  float P = mob_apply(excl, scal[2]);   // p0 through all preceding blocks
  for (int i = 0; i < g; ++i) {
    int b = k * g + i;
    if (b < nb) { pcar[b] = P; P = mob_apply(mob_ld(blkM, b), P); }
  }
}

// ---------------- K5: fill P_filt + per-block affine aggregates for m ----------------
__global__ void ks_fwdP_fill(const float* __restrict__ times, const float* __restrict__ values,
                             int n, const float* __restrict__ scal, const float* __restrict__ pcar,
                             float* __restrict__ Pf, float* __restrict__ blkA) {
  extern __shared__ float sm[];
  float* sT = sm;             // TILE
  float* sV = sm + TILE;      // TILE
  float* sS = sm + 2 * TILE;  // NT*4 (reused NT*2 for affine)
  int lo = (int)blockIdx.x * TILE;
  int hi = lo + TILE; if (hi > n) hi = n;
  stage_to_lds(times + lo, sT, hi - lo);
  stage_to_lds(values + lo, sV, hi - lo);
  stage_wait();
  float qb = scal[0], rr = scal[1];
  int k = (int)threadIdx.x;
  int s = lo + k * PER;
  int e = s + PER; if (e > hi) e = hi;
  Mob acc = thread_mob_agg(times, sT, lo, hi, s, e, qb, rr);
  mob_scan_inc(sS, k, acc);
  Mob excl = (k == 0) ? mob_id() : mob_ld(sS, k - 1);
  float P = mob_apply(excl, pcar[blockIdx.x]);
  Aff am = aff_id();
  float pbuf[PER];
  int cnt = (s < hi) ? (e - s) : 0;
  if (cnt > 0) {
    float tprev = 0.f;
    if (s > 0) tprev = ld_gs(times, sT, lo, hi, s - 1);
    for (int t = s; t < e; ++t) {
      float tc = sT[t - lo];
      float dtv = (t == 0) ? 1.0f : fmaxf(tc - tprev, MINVARF);
      tprev = tc;
      float q = qb * dtv;
      float Pp = P + q;
      float S = Pp + rr;
      float K = Pp / S;
      float Pfv = (1.f - K) * Pp;
      pbuf[t - s] = Pfv;
      P = Pfv;
      Aff f; f.a = 1.f - K; f.b = K * sV[t - lo];
      am = aff_comp(f, am);
    }
    for (int j = 0; j < cnt; ++j) Pf[s + j] = pbuf[j];
  }
  __syncthreads();  // done reading Moebius scan data before reuse
  aff_scan_inc(sS, k, am);
  if (k == NT - 1) aff_st(blkA, (int)blockIdx.x, aff_ld(sS, k));
}

// ---------------- K6: m carry-in per block ----------------
__global__ void ks_carryM(const float* __restrict__ blkA, int nb,
                          const float* __restrict__ values, float* __restrict__ mcar) {
  __shared__ float sS[NT * 2];
  int k = (int)threadIdx.x;
  int g = (nb + NT - 1) / NT;
  Aff acc = aff_id();
  for (int i = 0; i < g; ++i) { int b = k * g + i; if (b < nb) acc = aff_comp(aff_ld(blkA, b), acc); }
  aff_scan_inc(sS, k, acc);
  Aff excl = (k == 0) ? aff_id() : aff_ld(sS, k - 1);
  float m = excl.a * values[0] + excl.b;   // carry init m_prev = values[0]
  for (int i = 0; i < g; ++i) {
    int b = k * g + i;
    if (b < nb) { mcar[b] = m; Aff f = aff_ld(blkA, b); m = f.a * m + f.b; }
  }
}

// ---------------- K7: fill m_filt into d_out ----------------
__global__ void ks_fwdM_fill(const float* __restrict__ times, const float* __restrict__ values,
                             int n, const float* __restrict__ scal, const float* __restrict__ Pf,
                             const float* __restrict__ mcar, float* __restrict__ out) {
  extern __shared__ float sm[];
  float* sT = sm;
  float* sV = sm + TILE;
  float* sP = sm + 2 * TILE;
  float* sS = sm + 3 * TILE;  // NT*2
  int lo = (int)blockIdx.x * TILE;
  int hi = lo + TILE; if (hi > n) hi = n;
  stage_to_lds(times + lo, sT, hi - lo);
  stage_to_lds(values + lo, sV, hi - lo);
  stage_to_lds(Pf + lo, sP, hi - lo);
  stage_wait();
  float qb = scal[0], rr = scal[1], p0 = scal[2];
  int k = (int)threadIdx.x;
  int s = lo + k * PER;
  int e = s + PER; if (e > hi) e = hi;
  int cnt = (s < hi) ? (e - s) : 0;
  float aa[PER], bb[PER];
  Aff acc = aff_id();
  if (cnt > 0) {
    float tprev = 0.f;
    if (s > 0) tprev = ld_gs(times, sT, lo, hi, s - 1);
    float pprev = (s == 0) ? p0 : ld_gs(Pf, sP, lo, hi, s - 1);
    for (int t = s; t < e; ++t) {
      float tc = sT[t - lo];
      float dtv = (t == 0) ? 1.0f : fmaxf(tc - tprev, MINVARF);
      tprev = tc;
      float q = qb * dtv;
      float a = rr / (pprev + q + rr);        // 1-K = r/S
      float b = (1.f - a) * sV[t - lo];       // K*y
      aa[t - s] = a; bb[t - s] = b;
      Aff f; f.a = a; f.b = b;
      acc = aff_comp(f, acc);
      pprev = sP[t - lo];
    }
  }
  aff_scan_inc(sS, k, acc);
  Aff excl = (k == 0) ? aff_id() : aff_ld(sS, k - 1);
  float m = excl.a * mcar[blockIdx.x] + excl.b;
  for (int j = 0; j < cnt; ++j) { m = aa[j] * m + bb[j]; out[s + j] = m; }
}

// ---------------- K8: backward (RTS) per-block aggregates ----------------
__global__ void ks_bwd_agg(const float* __restrict__ times, int n,
                           const float* __restrict__ scal, const float* __restrict__ Pf,
                           const float* __restrict__ mf, float* __restrict__ blkB) {
  extern __shared__ float sm[];
  float* sT = sm;
  float* sP = sm + TILE;
  float* sM = sm + 2 * TILE;
  float* sS = sm + 3 * TILE;  // NT*2
  int lo = (int)blockIdx.x * TILE;
  int hi = lo + TILE; if (hi > n) hi = n;
  stage_to_lds(times + lo, sT, hi - lo);
  stage_to_lds(Pf + lo, sP, hi - lo);
  stage_to_lds(mf + lo, sM, hi - lo);
  stage_wait();
  float qb = scal[0];
  int k = (int)threadIdx.x;
  int s = lo + k * PER;
  int e = s + PER; if (e > hi) e = hi;
  Aff acc = aff_id();
  if (s < hi) {
    for (int t = e - 1; t >= s; --t) {
      float a, b;
      if (t == n - 1) { a = 0.f; b = sM[t - lo]; }  // ms[n-1] = m_filt[n-1]
      else {
        float tn = ld_gs(times, sT, lo, hi, t + 1);
        float tc = sT[t - lo];
        float dtv = fmaxf(tn - tc, MINVARF);
        float q1 = qb * dtv;
        float pf = sP[t - lo];
        float J = pf / (pf + q1);                  // P_f[t]/P_pred[t+1]
        a = J; b = (1.f - J) * sM[t - lo];
      }
      Aff f; f.a = a; f.b = b;
      acc = aff_comp(f, acc);
    }
  }
  aff_scan_rev(sS, k, acc);
  if (k == 0) aff_st(blkB, (int)blockIdx.x, aff_ld(sS, 0));
}

// ---------------- K9: backward carries (entering each block from the right) ----------------
__global__ void ks_carryB(const float* __restrict__ blkB, int nb, float* __restrict__ bcar) {
  __shared__ float sS[NT * 2];
  int k = (int)threadIdx.x;
  int g = (nb + NT - 1) / NT;
  Aff acc = aff_id();
  for (int i = g - 1; i >= 0; --i) { int b = k * g + i; if (b < nb) acc = aff_comp(aff_ld(blkB, b), acc); }
  aff_scan_rev(sS, k, acc);
  Aff excl = (k == NT - 1) ? aff_id() : aff_ld(sS, k + 1);
  float x = excl.b;  // applied to dummy 0 (killed by a=0 at t=n-1)
  for (int i = g - 1; i >= 0; --i) {
    int b = k * g + i;
    if (b < nb) { bcar[b] = x; Aff f = aff_ld(blkB, b); x = f.a * x + f.b; }
  }
}

// ---------------- K10: backward fill, in-place over d_out ----------------
__global__ void ks_bwd_fill(const float* __restrict__ times, int n,
                            const float* __restrict__ scal, const float* __restrict__ Pf,
                            float* __restrict__ mf, const float* __restrict__ bcar) {
  extern __shared__ float sm[];
  float* sT = sm;
  float* sP = sm + TILE;
  float* sM = sm + 2 * TILE;
  float* sS = sm + 3 * TILE;  // NT*2
  int lo = (int)blockIdx.x * TILE;
  int hi = lo + TILE; if (hi > n) hi = n;
  stage_to_lds(times + lo, sT, hi - lo);
  stage_to_lds(Pf + lo, sP, hi - lo);
  stage_to_lds(mf + lo, sM, hi - lo);
  stage_wait();
  float qb = scal[0];
  int k = (int)threadIdx.x;
  int s = lo + k * PER;
  int e = s + PER; if (e > hi) e = hi;
  int cnt = (s < hi) ? (e - s) : 0;
  float aa[PER], bb[PER];
  Aff acc = aff_id();
  if (cnt > 0) {
    for (int t = e - 1; t >= s; --t) {
      float a, b;
      if (t == n - 1) { a = 0.f; b = sM[t - lo]; }
      else {
        float tn = ld_gs(times, sT, lo, hi, t + 1);
        float tc = sT[t - lo];
        float dtv = fmaxf(tn - tc, MINVARF);
        float q1 = qb * dtv;
        float pf = sP[t - lo];
        float J = pf / (pf + q1);
        a = J; b = (1.f - J) * sM[t - lo];
      }
      aa[t - s] = a; bb[t - s] = b;
      Aff f; f.a = a; f.b = b;
      acc = aff_comp(f, acc);
    }
  }
  aff_scan_rev(sS, k, acc);
  Aff excl = (k == NT - 1) ? aff_id() : aff_ld(sS, k + 1);
  float x = excl.a * bcar[blockIdx.x] + excl.b;
  for (int j = cnt - 1; j >= 0; --j) { x = aa[j] * x + bb[j]; mf[s + j] = x; }  // in-place ms
}

// ---------------- host ----------------
extern "C" void kernel_launch(void* const* d_in, const int* in_sizes, int n_in,
                              void* d_out, int out_size, void* d_ws, size_t ws_size,
                              hipStream_t stream) {
  (void)n_in; (void)out_size; (void)ws_size;
  const float* times = (const float*)d_in[0];
  const float* values = (const float*)d_in[1];
  float* out = (float*)d_out;
  int n = in_sizes[0];
  int nb = (n + TILE - 1) / TILE;

  char* ws = (char*)d_ws;
  size_t off = 0;
  double* part = (double*)(ws + off); off += (size_t)NRED * 2 * sizeof(double);
  float* scal = (float*)(ws + off);   off += 256;
  float* blkM = (float*)(ws + off);   off += (size_t)nb * 4 * sizeof(float);
  float* pcar = (float*)(ws + off);   off += (size_t)nb * sizeof(float);
  float* blkA = (float*)(ws + off);   off += (size_t)nb * 2 * sizeof(float);
  float* mcar = (float*)(ws + off);   off += (size_t)nb * sizeof(float);
  float* blkB = (float*)(ws + off);   off += (size_t)nb * 2 * sizeof(float);
  float* bcar = (float*)(ws + off);   off += (size_t)nb * sizeof(float);
  off = (off + 255) & ~(size_t)255;
  float* Pf = (float*)(ws + off);     off += (size_t)n * sizeof(float);

  const size_t sm_agg = (size_t)(TILE + NT * 4) * sizeof(float);       // 20 KB
  const size_t sm_pf  = (size_t)(2 * TILE + NT * 4) * sizeof(float);   // 36 KB
  const size_t sm_big = (size_t)(3 * TILE + NT * 2) * sizeof(float);   // 50 KB

  ks_var_partial<<<NRED, NT, 0, stream>>>(values, n, part);
  ks_var_final<<<1, NT, 0, stream>>>(part, NRED, n, scal);
  ks_fwdP_agg<<<nb, NT, sm_agg, stream>>>(times, n, scal, blkM);
  ks_carryP<<<1, NT, 0, stream>>>(blkM, nb, scal, pcar);
  ks_fwdP_fill<<<nb, NT, sm_pf, stream>>>(times, values, n, scal, pcar, Pf, blkA);
  ks_carryM<<<1, NT, 0, stream>>>(blkA, nb, values, mcar);
  ks_fwdM_fill<<<nb, NT, sm_big, stream>>>(times, values, n, scal, Pf, mcar, out);
  ks_bwd_agg<<<nb, NT, sm_big, stream>>>(times, n, scal, Pf, out, blkB);
  ks_carryB<<<1, NT, 0, stream>>>(blkB, nb, bcar);
  ks_bwd_fill<<<nb, NT, sm_big, stream>>>(times, n, scal, Pf, out, bcar);
}